// ConnectNet_65489661330159
// MI455X (gfx1250) — compile-verified
//
#include <hip/hip_runtime.h>

// CDNA5 / gfx1250 implicit-GEMM 3x3 conv using V_WMMA_F32_16X16X4_F32.
// M=16 (C_out), K=144 = (kh*3+kw)*16 + ci, N = spatial (64 px per wave).
// - K-loop: 3 kh iterations (#pragma unroll 1) x 12 unrolled steps so live
//   VGPRs stay < 256 while keeping kw-direction global_load_b96 merging.
// - B addressing via uniform base + per-lane i32 element offsets so the
//   compiler can use saddr+voffset32 loads (1 VALU add per load instead of
//   a 64-bit carry-chain pair).

typedef __attribute__((ext_vector_type(2))) float v2f;
typedef __attribute__((ext_vector_type(8))) float v8f;

#define H_IN   2048
#define W_IN   2048
#define C_IN   16
#define C_OUT  16
#define H_OUT  2046
#define W_OUT  2046
#define HWIN   (H_IN * W_IN)          // elements per input channel plane

__global__ __launch_bounds__(256) void conv3x3_wmma_f32(
    const float* __restrict__ x,      // (16, 2048, 2048)
    const float* __restrict__ wts,    // (16, 16, 3, 3) OIHW
    float* __restrict__ out)          // (16, 2046, 2046)
{
    // ---- Stage weights into LDS in [k][m] order, k = (kh*3+kw)*16 + ci ----
    __shared__ float ws[144 * 16];    // 9216 B
    const int tid = threadIdx.x;
    #pragma unroll
    for (int it = 0; it < 9; ++it) {  // 9 * 256 == 2304, exact: no divergence
        int g    = it * 256 + tid;    // g = ((m*16 + ci)*9 + khkw)
        int m    = g / 144;
        int r    = g - m * 144;
        int ci   = r / 9;
        int khkw = r - ci * 9;
        ws[(khkw * 16 + ci) * 16 + m] = wts[g];
    }
    __syncthreads();

    const int wave = tid >> 5;
    const int lane = tid & 31;
    const int n    = lane & 15;       // N index within 16-wide tile / M row for A
    const int half = lane >> 4;       // 0: lanes 0-15, 1: lanes 16-31

    const int oy = blockIdx.y;                        // output row, 0..2045
    const int xt = (blockIdx.x * 8 + wave) * 64;      // wave's output-x base

    // Per-lane i32 element offsets for the 4 N-tiles:
    //   offs[t] = (2*half)*HWIN + oy*W + clamp(x)   (b.x plane; b.y adds HWIN)
    // Clamp x to 2045 so x+kw <= 2047 stays in-bounds for edge tiles.
    const int obase = (2 * half) * HWIN + oy * W_IN;
    int offs[4];
    #pragma unroll
    for (int t = 0; t < 4; ++t) {
        int xx = xt + t * 16 + n;
        if (xx > W_IN - 3) xx = W_IN - 3;
        offs[t] = obase + xx;
    }

    // Per-lane LDS base for A fragments: element [k = 2*half][m = n].
    const int lds_lane = (2 * half) * 16 + n;

    v8f acc0 = {}, acc1 = {}, acc2 = {}, acc3 = {};

    // ---- Main K loop: 3 kh iterations x (3 kw x 4 ci-groups) x 4 tiles ----
    #pragma unroll 1
    for (int kh = 0; kh < 3; ++kh) {
        const int koff   = kh * W_IN;                 // input row oy + kh
        const int lds_kh = (kh * 3) * 64 + lds_lane;

        // Per-tile offsets for this input row (4 x v_add_nc_u32 per iter).
        int o0 = offs[0] + koff;
        int o1 = offs[1] + koff;
        int o2 = offs[2] + koff;
        int o3 = offs[3] + koff;

        #pragma unroll
        for (int kw = 0; kw < 3; ++kw) {
            #pragma unroll
            for (int cg = 0; cg < 4; ++cg) {
                const int ci0 = cg * 4;
                const int cst = ci0 * HWIN + kw;      // compile-time constant

                // A fragment from LDS: k = (kh*3+kw)*4 + ci0 + 2*half (+1)
                const int la = lds_kh + kw * 64 + ci0 * 16;
                v2f a;
                a.x = ws[la];
                a.y = ws[la + 16];

                v2f b0, b1, b2, b3;
                b0.x = x[o0 + cst];        b0.y = x[o0 + cst + HWIN];
                b1.x = x[o1 + cst];        b1.y = x[o1 + cst + HWIN];
                b2.x = x[o2 + cst];        b2.y = x[o2 + cst + HWIN];
                b3.x = x[o3 + cst];        b3.y = x[o3 + cst + HWIN];

                acc0 = __builtin_amdgcn_wmma_f32_16x16x4_f32(false, a, false, b0,
                                                             (short)0, acc0, false, false);
                acc1 = __builtin_amdgcn_wmma_f32_16x16x4_f32(false, a, false, b1,
                                                             (short)0, acc1, false, false);
                acc2 = __builtin_amdgcn_wmma_f32_16x16x4_f32(false, a, false, b2,
                                                             (short)0, acc2, false, false);
                acc3 = __builtin_amdgcn_wmma_f32_16x16x4_f32(false, a, false, b3,
                                                             (short)0, acc3, false, false);
            }
        }
    }

    // ---- Store D tiles. D layout: vgpr v -> M = v + 8*half, N = lane&15 ----
    const size_t plane = (size_t)H_OUT * W_OUT;
    float* op = out + (size_t)oy * W_OUT;
    v8f accs[4] = {acc0, acc1, acc2, acc3};
    #pragma unroll
    for (int t = 0; t < 4; ++t) {
        int ox = xt + t * 16 + n;
        if (ox < W_OUT) {
            #pragma unroll
            for (int v = 0; v < 8; ++v) {
                op[(size_t)(v + 8 * half) * plane + ox] = accs[t][v];
            }
        }
    }
}

extern "C" void kernel_launch(void* const* d_in, const int* in_sizes, int n_in,
                              void* d_out, int out_size, void* d_ws, size_t ws_size,
                              hipStream_t stream) {
    const float* x   = (const float*)d_in[0];   // (16, 2048, 2048) fp32
    const float* wts = (const float*)d_in[1];   // (16, 16, 3, 3)  fp32
    float* out       = (float*)d_out;           // (16, 2046, 2046) fp32

    // 4 blocks/row * 8 waves/block * 64 px/wave = 2048 >= 2046 output cols.
    dim3 grid(4, H_OUT);
    conv3x3_wmma_f32<<<grid, 256, 0, stream>>>(x, wts, out);
}